// PointPWC_65987877535945
// MI455X (gfx1250) — compile-verified
//
#include <hip/hip_runtime.h>

#define NPTS 8192
#define NSPLIT 8
#define TILES_PER_SPLIT (NPTS / 16 / NSPLIT)   // 64

typedef __attribute__((ext_vector_type(2))) float v2f;
typedef __attribute__((ext_vector_type(8))) float v8f;

#define F32_INF __uint_as_float(0x7F800000u)
// sentinel key: (flip(+INF) << 32) | all-ones  -> decodes to +INF (NOT NaN)
#define KEY_INF ((0xFF800000ULL << 32) | 0xFFFFFFFFULL)

// ---- order-preserving float<->uint mapping (branchless) ----
__device__ __forceinline__ unsigned flipf(float f) {
    unsigned b = __float_as_uint(f);
    return b ^ ((unsigned)((int)b >> 31) | 0x80000000u);
}
__device__ __forceinline__ float unflipf(unsigned u) {
    return __uint_as_float(u ^ ((unsigned)((int)(~u) >> 31) | 0x80000000u));
}
// sortable key: (flip(dist) << 32) | idx  -> ascending (dist, idx) order
__device__ __forceinline__ unsigned long long mkkey(float d, int j) {
    return ((unsigned long long)flipf(d) << 32) | (unsigned)j;
}

// ---- prep: build WMMA A-operand / B-operand forms + squared norms ----
// A row = (-2x, -2y, -2z, |p|^2) (K=0..3); B col = (x, y, z, 1); C = |q|^2
// => D = |p|^2 - 2 p.q + |q|^2
__global__ void prep_kernel(const float* __restrict__ pred,
                            const float* __restrict__ gt,
                            const float* __restrict__ coords,
                            float4* __restrict__ pc1_p, float4* __restrict__ pc1_a,
                            float4* __restrict__ pc2_p, float4* __restrict__ pc2_a,
                            float4* __restrict__ warp_p, float4* __restrict__ warp_a,
                            unsigned* __restrict__ colbits) {
    int i = blockIdx.x * blockDim.x + threadIdx.x;
    if (i >= NPTS) return;
    float cx = coords[3*i], cy = coords[3*i+1], cz = coords[3*i+2];
    float px = pred[3*i],   py = pred[3*i+1],   pz = pred[3*i+2];
    float gx = gt[3*i],     gy = gt[3*i+1],     gz = gt[3*i+2];

    float s1 = cx*cx + cy*cy + cz*cz;
    pc1_p[i] = make_float4(cx, cy, cz, s1);
    pc1_a[i] = make_float4(-2.f*cx, -2.f*cy, -2.f*cz, s1);

    float qx = cx + gx, qy = cy + gy, qz = cz + gz;
    float s2 = qx*qx + qy*qy + qz*qz;
    pc2_p[i] = make_float4(qx, qy, qz, s2);
    pc2_a[i] = make_float4(-2.f*qx, -2.f*qy, -2.f*qz, s2);

    float wx = cx + px, wy = cy + py, wz = cz + pz;
    float sw = wx*wx + wy*wy + wz*wz;
    warp_p[i] = make_float4(wx, wy, wz, sw);
    warp_a[i] = make_float4(-2.f*wx, -2.f*wy, -2.f*wz, sw);

    colbits[i] = flipf(F32_INF);
}

// ---- pure-u64 sorted top-K insertion (merge kernels) ----
template<int K>
__device__ __forceinline__ void topk_insert_key(unsigned long long (&kk)[K],
                                                unsigned long long nk) {
    if (nk < kk[K-1]) {
        kk[K-1] = nk;
        #pragma unroll
        for (int q = K-1; q > 0; --q) {
            unsigned long long x = kk[q], y = kk[q-1];
            bool sw = x < y;
            kk[q-1] = sw ? x : y;
            kk[q]   = sw ? y : x;
        }
    }
}

// ---- hot-path insertion: cheap float gate, key built only on accept ----
// Strict float '<' is stability-exact here: within a lane's stream the index is
// strictly increasing, so an equal-distance candidate ranks after the K-th entry.
template<int K>
__device__ __forceinline__ void topk_insert_cand(unsigned long long (&kk)[K],
                                                 float& kdLast, float dv, int j) {
    if (dv < kdLast) {                       // single v_cmp_lt_f32 gate
        kk[K-1] = mkkey(dv, j);
        #pragma unroll
        for (int q = K-1; q > 0; --q) {
            unsigned long long x = kk[q], y = kk[q-1];
            bool sw = x < y;
            kk[q-1] = sw ? x : y;            // native v_min_u64 / v_max_u64
            kk[q]   = sw ? y : x;
        }
        kdLast = unflipf((unsigned)(kk[K-1] >> 32));
    }
}

// ---- WMMA-tiled kNN pass: one wave per (16-row block, column split) ----
// Emits a sorted partial top-K key list per (row, split) into part_keys.
template<int K, bool CROSS>
__global__ void __launch_bounds__(32)
knn_pass_kernel(const float4* __restrict__ A4,      // A-form rows (src)
                const float4* __restrict__ B4,      // B-form cols (dst): (x,y,z,|q|^2)
                unsigned long long* __restrict__ part_keys,  // [NPTS][NSPLIT][K]
                unsigned* __restrict__ colmin) {    // [NPTS] (CROSS only)
    const int blk = blockIdx.x;
    const int rb  = blk >> 3;              // 16-row block (0..511)
    const int sp  = blk & (NSPLIT - 1);    // column split (0..7)
    const int l   = threadIdx.x;           // 0..31, wave32
    const int r0  = l & 15;
    const int h   = l >> 4;
    const int row = rb * 16 + r0;

    // A operand: lane holds row M=r0; halves hold K={0,1} / {2,3}
    float4 a4 = A4[row];
    v2f a; a.x = h ? a4.z : a4.x; a.y = h ? a4.w : a4.y;

    unsigned long long kk[K];
    #pragma unroll
    for (int q = 0; q < K; ++q) kk[q] = KEY_INF;
    float kdLast = F32_INF;

    __shared__ float tile[16 * 20];        // stride 20: conflict-free + 16B aligned rows
    __shared__ unsigned long long mk[16][2 * K];

    const int tbase = sp * TILES_PER_SPLIT;
    float4 p = B4[tbase * 16 + r0];        // software-pipelined B-point
    for (int tt = 0; tt < TILES_PER_SPLIT; ++tt) {
        const int t = tbase + tt;
        // prefetch next tile's B point (wraps harmlessly on last iter)
        float4 pn = B4[(tbase + ((tt + 1) & (TILES_PER_SPLIT - 1))) * 16 + r0];

        // B operand: lane holds col N=r0 of this tile
        v2f b; b.x = h ? p.z : p.x; b.y = h ? 1.0f : p.y;
        v8f cacc;
        #pragma unroll
        for (int r = 0; r < 8; ++r) cacc[r] = p.w;   // C = |q|^2

        v8f d = __builtin_amdgcn_wmma_f32_16x16x4_f32(
            false, a, false, b, (short)0, cacc, false, false);

        if (CROSS) {
            // all 8 elements in a lane share column N=r0 -> cheap col-min
            float cm = fminf(fminf(fminf(d[0], d[1]), fminf(d[2], d[3])),
                             fminf(fminf(d[4], d[5]), fminf(d[6], d[7])));
            atomicMin(&colmin[t * 16 + r0], flipf(cm));
        }

        // spill tile: lane l, reg r -> (M = r + 8h, N = r0)
        #pragma unroll
        for (int r = 0; r < 8; ++r) tile[(r + 8 * h) * 20 + r0] = d[r];
        __syncthreads();

        // transposed consume as 2x b128: lane handles row r0, cols [8h, 8h+8)
        const int cb = 8 * h;
        float4 c0 = *(const float4*)&tile[r0 * 20 + cb];
        float4 c1 = *(const float4*)&tile[r0 * 20 + cb + 4];
        // per-tile gate: skip all 8 candidates when none can enter the top-K
        float m8 = fminf(fminf(fminf(c0.x, c0.y), fminf(c0.z, c0.w)),
                         fminf(fminf(c1.x, c1.y), fminf(c1.z, c1.w)));
        if (m8 < kdLast) {
            const int gj = t * 16 + cb;
            topk_insert_cand<K>(kk, kdLast, c0.x, gj + 0);
            topk_insert_cand<K>(kk, kdLast, c0.y, gj + 1);
            topk_insert_cand<K>(kk, kdLast, c0.z, gj + 2);
            topk_insert_cand<K>(kk, kdLast, c0.w, gj + 3);
            topk_insert_cand<K>(kk, kdLast, c1.x, gj + 4);
            topk_insert_cand<K>(kk, kdLast, c1.y, gj + 5);
            topk_insert_cand<K>(kk, kdLast, c1.z, gj + 6);
            topk_insert_cand<K>(kk, kdLast, c1.w, gj + 7);
        }
        __syncthreads();
        p = pn;
    }

    // merge the two per-row lane lists (keys are unique -> plain u64 compare)
    #pragma unroll
    for (int q = 0; q < K; ++q) mk[r0][h * K + q] = kk[q];
    __syncthreads();
    if (h == 0) {
        unsigned long long* dst = &part_keys[(size_t)(row * NSPLIT + sp) * K];
        int p0 = 0, p1 = 0;
        for (int q = 0; q < K; ++q) {
            unsigned long long k0 = mk[r0][p0], k1 = mk[r0][K + p1];
            bool t0 = k0 < k1;
            dst[q] = t0 ? k0 : k1;
            p0 += t0; p1 += !t0;
        }
    }
}

// ---- merge NSPLIT sorted partial lists -> final top-K per row ----
template<int K, bool CROSS>
__global__ void knn_merge_kernel(const unsigned long long* __restrict__ part_keys,
                                 int* __restrict__ idx_out,
                                 float* __restrict__ dist_out) {
    int i = blockIdx.x * blockDim.x + threadIdx.x;
    if (i >= NPTS) return;
    unsigned long long best[K];
    #pragma unroll
    for (int q = 0; q < K; ++q) best[q] = KEY_INF;
    for (int s = 0; s < NSPLIT; ++s) {
        const unsigned long long* src = &part_keys[(size_t)(i * NSPLIT + s) * K];
        #pragma unroll
        for (int q = 0; q < K; ++q) topk_insert_key<K>(best, src[q]);
    }
    #pragma unroll
    for (int q = 0; q < K; ++q) {
        idx_out[i * K + q] = (int)(unsigned)(best[q] & 0xFFFFFFFFu);
        if (CROSS) dist_out[i * K + q] = unflipf((unsigned)(best[q] >> 32));
    }
}

// ---- pc2 curvature: (sum_{k=10} pc2[knn] - 10*pc2[i]) / 9 ----
__global__ void curv2_kernel(const float4* __restrict__ pc2_p,
                             const int* __restrict__ knn2,
                             float4* __restrict__ curv_out) {
    int i = blockIdx.x * blockDim.x + threadIdx.x;
    if (i >= NPTS) return;
    float4 p = pc2_p[i];
    float sx = 0.f, sy = 0.f, sz = 0.f;
    #pragma unroll
    for (int q = 0; q < 10; ++q) {
        float4 g = pc2_p[knn2[i * 10 + q]];
        sx += g.x; sy += g.y; sz += g.z;
    }
    const float inv9 = 1.0f / 9.0f;
    curv_out[i] = make_float4((sx - 10.f * p.x) * inv9,
                              (sy - 10.f * p.y) * inv9,
                              (sz - 10.f * p.z) * inv9, 0.f);
}

// ---- per-row terms: moved curvature, smoothness, interpolated-curvature loss ----
__global__ void finalize_kernel(const float4* __restrict__ warp_p,
                                const int* __restrict__ knn1,
                                const float* __restrict__ flow,
                                const float4* __restrict__ curv2,
                                const int* __restrict__ cross_idx,
                                const float* __restrict__ cross_dist,
                                float* __restrict__ smooth_row,
                                float* __restrict__ curv_row) {
    int i = blockIdx.x * blockDim.x + threadIdx.x;
    if (i >= NPTS) return;

    int idx10[10];
    #pragma unroll
    for (int q = 0; q < 10; ++q) idx10[q] = knn1[i * 10 + q];

    float4 w = warp_p[i];
    float sx = 0.f, sy = 0.f, sz = 0.f;
    #pragma unroll
    for (int q = 0; q < 10; ++q) {
        float4 g = warp_p[idx10[q]];
        sx += g.x; sy += g.y; sz += g.z;
    }
    const float inv9 = 1.0f / 9.0f;
    float mcx = (sx - 10.f * w.x) * inv9;
    float mcy = (sy - 10.f * w.y) * inv9;
    float mcz = (sz - 10.f * w.z) * inv9;

    // smoothness: kNN(9) = first 9 of kNN(10); zero-safe norm
    float fx = flow[3*i], fy = flow[3*i+1], fz = flow[3*i+2];
    float s = 0.f;
    #pragma unroll
    for (int q = 0; q < 9; ++q) {
        int j = idx10[q];
        float dx = flow[3*j]   - fx;
        float dy = flow[3*j+1] - fy;
        float dz = flow[3*j+2] - fz;
        float sq = dx*dx + dy*dy + dz*dz;
        s += (sq > 0.f) ? sqrtf(sq) : 0.f;
    }
    smooth_row[i] = s * 0.125f;

    // inverse-distance interpolation of pc2 curvature at top-5 cross neighbors
    float wsum = 0.f, ax = 0.f, ay = 0.f, az = 0.f;
    #pragma unroll
    for (int q = 0; q < 5; ++q) {
        float dd = cross_dist[i * 5 + q];
        float wq = 1.0f / (dd + 1e-8f);
        float4 c = curv2[cross_idx[i * 5 + q]];
        wsum += wq; ax += wq * c.x; ay += wq * c.y; az += wq * c.z;
    }
    float inv = 1.0f / wsum;
    float dx = ax * inv - mcx, dy = ay * inv - mcy, dz = az * inv - mcz;
    curv_row[i] = dx*dx + dy*dy + dz*dz;
}

// ---- deterministic final reduction (single block, fixed order) ----
__global__ void reduce_kernel(const float* __restrict__ cross_dist,
                              const unsigned* __restrict__ colbits,
                              const float* __restrict__ smooth_row,
                              const float* __restrict__ curv_row,
                              float* __restrict__ out) {
    __shared__ float sa[256], sb[256], sc[256], sd[256];
    int tid = threadIdx.x;
    float a = 0.f, b = 0.f, c = 0.f, d = 0.f;
    for (int i = tid; i < NPTS; i += 256) {
        a += cross_dist[i * 5];       // dist1 = row-min (top-5[0])
        b += unflipf(colbits[i]);     // dist2 = col-min
        c += smooth_row[i];
        d += curv_row[i];
    }
    sa[tid] = a; sb[tid] = b; sc[tid] = c; sd[tid] = d;
    __syncthreads();
    for (int s2 = 128; s2 > 0; s2 >>= 1) {
        if (tid < s2) {
            sa[tid] += sa[tid + s2]; sb[tid] += sb[tid + s2];
            sc[tid] += sc[tid + s2]; sd[tid] += sd[tid + s2];
        }
        __syncthreads();
    }
    if (tid == 0) {
        // 0.02*chamfer + 0.3*0.02*curv + 1.0*0.02*smooth
        out[0] = 0.02f * (sa[0] + sb[0]) + 0.006f * sd[0] + 0.02f * sc[0];
    }
}

extern "C" void kernel_launch(void* const* d_in, const int* in_sizes, int n_in,
                              void* d_out, int out_size, void* d_ws, size_t ws_size,
                              hipStream_t stream) {
    (void)in_sizes; (void)n_in; (void)out_size; (void)ws_size;
    const float* pred   = (const float*)d_in[0];  // registration_pred [1,N,3]
    const float* gt     = (const float*)d_in[1];  // registration_gt   [1,N,3]
    const float* coords = (const float*)d_in[2];  // coords            [N,3]

    char* ws = (char*)d_ws;
    size_t off = 0;
    auto wsalloc = [&](size_t bytes) -> void* {
        void* p = (void*)(ws + off);
        off += (bytes + 255) & ~(size_t)255;
        return p;
    };
    float4*   pc1_p      = (float4*)  wsalloc(NPTS * sizeof(float4));
    float4*   pc1_a      = (float4*)  wsalloc(NPTS * sizeof(float4));
    float4*   pc2_p      = (float4*)  wsalloc(NPTS * sizeof(float4));
    float4*   pc2_a      = (float4*)  wsalloc(NPTS * sizeof(float4));
    float4*   warp_p     = (float4*)  wsalloc(NPTS * sizeof(float4));
    float4*   warp_a     = (float4*)  wsalloc(NPTS * sizeof(float4));
    float4*   curv2      = (float4*)  wsalloc(NPTS * sizeof(float4));
    int*      knn1       = (int*)     wsalloc(NPTS * 10 * sizeof(int));
    int*      knn2       = (int*)     wsalloc(NPTS * 10 * sizeof(int));
    int*      cross_idx  = (int*)     wsalloc(NPTS * 5 * sizeof(int));
    float*    cross_dist = (float*)   wsalloc(NPTS * 5 * sizeof(float));
    unsigned* colbits    = (unsigned*)wsalloc(NPTS * sizeof(unsigned));
    float*    smooth_row = (float*)   wsalloc(NPTS * sizeof(float));
    float*    curv_row   = (float*)   wsalloc(NPTS * sizeof(float));
    unsigned long long* part_keys =
        (unsigned long long*)wsalloc((size_t)NPTS * NSPLIT * 10 * sizeof(unsigned long long));
    // total ~7.2 MB of workspace

    prep_kernel<<<NPTS / 256, 256, 0, stream>>>(pred, gt, coords,
        pc1_p, pc1_a, pc2_p, pc2_a, warp_p, warp_a, colbits);

    const int NPASS = (NPTS / 16) * NSPLIT;  // 4096 single-wave blocks per pass

    // self-kNN(10) on pc1
    knn_pass_kernel<10, false><<<NPASS, 32, 0, stream>>>(pc1_a, pc1_p, part_keys, nullptr);
    knn_merge_kernel<10, false><<<NPTS / 256, 256, 0, stream>>>(part_keys, knn1, nullptr);
    // self-kNN(10) on pc2
    knn_pass_kernel<10, false><<<NPASS, 32, 0, stream>>>(pc2_a, pc2_p, part_keys, nullptr);
    knn_merge_kernel<10, false><<<NPTS / 256, 256, 0, stream>>>(part_keys, knn2, nullptr);
    // cross pass: top-5 per row + column min (chamfer dist2)
    knn_pass_kernel<5, true><<<NPASS, 32, 0, stream>>>(warp_a, pc2_p, part_keys, colbits);
    knn_merge_kernel<5, true><<<NPTS / 256, 256, 0, stream>>>(part_keys, cross_idx, cross_dist);

    curv2_kernel<<<NPTS / 256, 256, 0, stream>>>(pc2_p, knn2, curv2);
    finalize_kernel<<<NPTS / 256, 256, 0, stream>>>(warp_p, knn1, pred, curv2,
        cross_idx, cross_dist, smooth_row, curv_row);
    reduce_kernel<<<1, 256, 0, stream>>>(cross_dist, colbits, smooth_row, curv_row,
        (float*)d_out);
}